// SupportGAT_386547057270
// MI455X (gfx1250) — compile-verified
//
#include <hip/hip_runtime.h>
#include <hip/hip_bf16.h>
#include <math.h>

typedef __attribute__((ext_vector_type(2))) float v2f;
typedef __attribute__((ext_vector_type(8))) float v8f;

#define NEG_SLOPE 0.2f

// ---------------------------------------------------------------------------
// FP32 GEMM via V_WMMA_F32_16X16X4_F32.  C[M,N] = A[M,K] @ B[K,N], row-major.
// One wave computes one 16x16 tile; K stepped by 4.
// A frag (16x4 f32): lanes 0-15 -> rows m0+l, VGPR0/1 = K 0/1; lanes 16-31 = K 2/3.
// B frag (4x16 f32): mirrored (rows=K striped across halves, cols across lanes).
// C/D (16x16 f32):   VGPR r -> row m0 + r + 8*half, col n0 + (lane&15).
// ---------------------------------------------------------------------------
__global__ void __launch_bounds__(128)
gemm_f32_wmma(const float* __restrict__ A, const float* __restrict__ B,
              float* __restrict__ C, int M, int N, int K) {
  const int wave = threadIdx.x >> 5;
  const int lane = threadIdx.x & 31;
  const int half = lane >> 4;
  const int l    = lane & 15;
  const int tilesN = N >> 4;
  const int tilesM = (M + 15) >> 4;
  const int tile = blockIdx.x * 4 + wave;          // uniform per wave
  if (tile >= tilesM * tilesN) return;             // whole-wave exit: EXEC stays full
  const int m0 = (tile / tilesN) << 4;
  const int n0 = (tile % tilesN) << 4;

  int arow = m0 + l;
  if (arow >= M) arow = M - 1;                     // clamp (store is guarded)
  const float* __restrict__ Arow = A + (size_t)arow * K;
  const float* __restrict__ Bp   = B + n0 + l;

  v8f acc = {};
  for (int k = 0; k < K; k += 4) {
    v2f a = *(const v2f*)(Arow + k + 2 * half);    // 8B-aligned contiguous pair
    v2f b;
    b.x = Bp[(size_t)(k + 2 * half) * N];
    b.y = Bp[(size_t)(k + 2 * half + 1) * N];
    acc = __builtin_amdgcn_wmma_f32_16x16x4_f32(false, a, false, b,
                                                (short)0, acc, false, false);
  }
#pragma unroll
  for (int r = 0; r < 8; ++r) {
    const int row = m0 + r + 8 * half;
    if (row < M) C[(size_t)row * N + n0 + l] = acc[r];
  }
}

// ---------------------------------------------------------------------------
// Per-node attention logits: alpha_src[n,h] = sum_d h[n,h,d]*a_src[h,d] (same dst)
// One wave per node; lanes stride D, wave-reduce with shfl_xor (wave32).
// ---------------------------------------------------------------------------
__global__ void alphas_kernel(const float* __restrict__ h,
                              const float* __restrict__ a_src,
                              const float* __restrict__ a_dst,
                              float* __restrict__ alpha_src,
                              float* __restrict__ alpha_dst,
                              int N, int H, int D) {
  const int wv   = (int)((blockIdx.x * blockDim.x + threadIdx.x) >> 5);
  const int lane = threadIdx.x & 31;
  if (wv >= N) return;
  for (int hh = 0; hh < H; ++hh) {
    float ss = 0.f, sd = 0.f;
    const float* row = h + (size_t)wv * H * D + (size_t)hh * D;
    for (int j = lane; j < D; j += 32) {
      const float v = row[j];
      ss += v * a_src[hh * D + j];
      sd += v * a_dst[hh * D + j];
    }
    for (int off = 16; off; off >>= 1) {
      ss += __shfl_xor(ss, off, 32);
      sd += __shfl_xor(sd, off, 32);
    }
    if (lane == 0) {
      alpha_src[wv * H + hh] = ss;
      alpha_dst[wv * H + hh] = sd;
    }
  }
}

// monotonic float<->uint map so unsigned atomicMax == float max
__device__ __forceinline__ unsigned fmap(float f) {
  unsigned u = __float_as_uint(f);
  return (u & 0x80000000u) ? ~u : (u | 0x80000000u);
}
__device__ __forceinline__ float funmap(unsigned u) {
  unsigned b = (u & 0x80000000u) ? (u & 0x7FFFFFFFu) : ~u;
  return __uint_as_float(b);
}
__device__ __forceinline__ void edge_sd(const int* __restrict__ ei, int e, int E,
                                        int& s, int& d) {
  if (e < E) { s = ei[e]; d = ei[E + e]; }
  else       { s = e - E; d = e - E; }            // appended self-loops
}
__device__ __forceinline__ float leaky(float v) {
  return v > 0.f ? v : NEG_SLOPE * v;
}

// Pass 1: segment max over dst (per head)
__global__ void edge_max_kernel(const int* __restrict__ ei, int E, int N,
                                const float* __restrict__ asrc,
                                const float* __restrict__ adst,
                                unsigned* __restrict__ m, int H) {
  const int e = blockIdx.x * blockDim.x + threadIdx.x;
  if (e >= E + N) return;
  int s, d; edge_sd(ei, e, E, s, d);
  for (int hh = 0; hh < H; ++hh) {
    const float v = leaky(asrc[s * H + hh] + adst[d * H + hh]);
    atomicMax(&m[d * H + hh], fmap(v));
  }
}

// Pass 2: sum of exp(e - m[dst])
__global__ void edge_expsum_kernel(const int* __restrict__ ei, int E, int N,
                                   const float* __restrict__ asrc,
                                   const float* __restrict__ adst,
                                   const unsigned* __restrict__ m,
                                   float* __restrict__ ssum, int H) {
  const int e = blockIdx.x * blockDim.x + threadIdx.x;
  if (e >= E + N) return;
  int s, d; edge_sd(ei, e, E, s, d);
  for (int hh = 0; hh < H; ++hh) {
    const float v = leaky(asrc[s * H + hh] + adst[d * H + hh]);
    atomicAdd(&ssum[d * H + hh], expf(v - funmap(m[d * H + hh])));
  }
}

// Pass 3: agg[dst,h,:] += h[src,h,:] * softmax_weight.  One wave per edge.
__global__ void edge_agg_kernel(const int* __restrict__ ei, int E, int N,
                                const float* __restrict__ h,
                                const float* __restrict__ asrc,
                                const float* __restrict__ adst,
                                const unsigned* __restrict__ m,
                                const float* __restrict__ ssum,
                                float* __restrict__ agg, int H, int D) {
  const int wv   = (int)((blockIdx.x * blockDim.x + threadIdx.x) >> 5);
  const int lane = threadIdx.x & 31;
  if (wv >= E + N) return;
  int s, d; edge_sd(ei, wv, E, s, d);
  for (int hh = 0; hh < H; ++hh) {
    const float v = leaky(asrc[s * H + hh] + adst[d * H + hh]);
    const float a = expf(v - funmap(m[d * H + hh])) /
                    fmaxf(ssum[d * H + hh], 1e-16f);
    const float* __restrict__ hrow = h   + (size_t)s * H * D + (size_t)hh * D;
    float* __restrict__       arow = agg + (size_t)d * H * D + (size_t)hh * D;
    for (int j = lane; j < D; j += 32)
      atomicAdd(&arow[j], hrow[j] * a);
  }
}

// mean over heads + bias + ELU
__global__ void finalize_kernel(const float* __restrict__ agg,
                                const float* __restrict__ b,
                                float* __restrict__ out, int N, int H, int D) {
  const int i = blockIdx.x * blockDim.x + threadIdx.x;
  if (i >= N * D) return;
  const int n = i / D, dd = i - n * D;
  float v = 0.f;
  for (int hh = 0; hh < H; ++hh) v += agg[(size_t)n * H * D + (size_t)hh * D + dd];
  v = v / (float)H + b[dd];
  out[i] = v > 0.f ? v : (expf(v) - 1.0f);
}

// global mean pool accumulation
__global__ void pool_kernel(const float* __restrict__ hfin,
                            const int* __restrict__ batch,
                            float* __restrict__ sums, float* __restrict__ cnt,
                            int N, int D) {
  const int i = blockIdx.x * blockDim.x + threadIdx.x;
  if (i >= N * D) return;
  const int n = i / D, dd = i - n * D;
  const int g = batch[n];
  atomicAdd(&sums[g * D + dd], hfin[i]);
  if (dd == 0) atomicAdd(&cnt[g], 1.0f);
}

// MLP head: one block (128 threads) per graph
__global__ void __launch_bounds__(128)
head_kernel(const float* __restrict__ sums, const float* __restrict__ cnt,
            const float* __restrict__ Wc1, const float* __restrict__ bc1,
            const float* __restrict__ Wc2, const float* __restrict__ bc2,
            float* __restrict__ out, int D) {
  __shared__ float pooled[128];
  __shared__ float red[128];
  const int g = blockIdx.x, j = threadIdx.x;
  const float c = fmaxf(cnt[g], 1.0f);
  pooled[j] = sums[g * D + j] / c;
  __syncthreads();
  float acc = bc1[j];
  for (int dd = 0; dd < D; ++dd) acc += pooled[dd] * Wc1[dd * D + j];
  const float z = fmaxf(acc, 0.0f);
  red[j] = z * Wc2[j];
  __syncthreads();
  for (int off = 64; off > 0; off >>= 1) {
    if (j < off) red[j] += red[j + off];
    __syncthreads();
  }
  if (j == 0) out[g] = red[0] + bc2[0];
}

// ---------------------------------------------------------------------------
extern "C" void kernel_launch(void* const* d_in, const int* in_sizes, int n_in,
                              void* d_out, int out_size, void* d_ws, size_t ws_size,
                              hipStream_t stream) {
  const float* x    = (const float*)d_in[0];
  const int*   ei   = (const int*)d_in[1];
  const int*   batch= (const int*)d_in[2];
  const float* W1   = (const float*)d_in[3];
  const float* aS1  = (const float*)d_in[4];
  const float* aD1  = (const float*)d_in[5];
  const float* b1   = (const float*)d_in[6];
  const float* W2   = (const float*)d_in[7];
  const float* aS2  = (const float*)d_in[8];
  const float* aD2  = (const float*)d_in[9];
  const float* b2   = (const float*)d_in[10];
  const float* Wc1  = (const float*)d_in[11];
  const float* bc1  = (const float*)d_in[12];
  const float* Wc2  = (const float*)d_in[13];
  const float* bc2  = (const float*)d_in[14];
  float* out = (float*)d_out;
  (void)n_in; (void)ws_size;

  const int FIN = 256, D = 128, H1 = 2;
  const int N    = in_sizes[0] / FIN;        // 50000
  const int E    = in_sizes[1] / 2;          // 800000
  const int Etot = E + N;                    // + self loops
  const int G    = out_size;                 // 64

  // ---- carve workspace (aligned to 256B) ----
  char* ws = (char*)d_ws; size_t off = 0;
  auto carve = [&](size_t bytes) -> void* {
    void* p = ws + off; off += (bytes + 255) & ~(size_t)255; return p;
  };
  float*    bufA = (float*)carve((size_t)N * FIN * 4);   // h1, later h2
  float*    bufB = (float*)carve((size_t)N * FIN * 4);   // agg1, later agg2
  float*    bufC = (float*)carve((size_t)N * D * 4);     // layer1 out, later layer2 out
  float*    as1  = (float*)carve((size_t)N * H1 * 4);
  float*    ad1  = (float*)carve((size_t)N * H1 * 4);
  unsigned* m1   = (unsigned*)carve((size_t)N * H1 * 4);
  float*    s1   = (float*)carve((size_t)N * H1 * 4);
  float*    as2  = (float*)carve((size_t)N * 4);
  float*    ad2  = (float*)carve((size_t)N * 4);
  unsigned* m2   = (unsigned*)carve((size_t)N * 4);
  float*    s2   = (float*)carve((size_t)N * 4);
  float*    sums = (float*)carve((size_t)G * D * 4);
  float*    cnt  = (float*)carve((size_t)G * 4);

  // ---- zero accumulators (graph-capture-safe) ----
  hipMemsetAsync(bufB, 0, (size_t)N * FIN * 4, stream);  // agg1
  hipMemsetAsync(m1,   0, (size_t)N * H1 * 4, stream);   // 0 < fmap(any float)
  hipMemsetAsync(s1,   0, (size_t)N * H1 * 4, stream);
  hipMemsetAsync(m2,   0, (size_t)N * 4, stream);
  hipMemsetAsync(s2,   0, (size_t)N * 4, stream);
  hipMemsetAsync(sums, 0, (size_t)G * D * 4, stream);
  hipMemsetAsync(cnt,  0, (size_t)G * 4, stream);

  const int edgeBlocks  = (Etot + 255) / 256;
  const int edgeWaveBlk = (Etot + 7) / 8;       // wave-per-edge kernels
  const int nodeWaveBlk = (N + 7) / 8;
  const int ndBlocks    = (N * D + 255) / 256;

  // ================= Layer 1 (heads=2, D=128) =================
  {
    const int tiles = ((N + 15) / 16) * (FIN / 16);
    gemm_f32_wmma<<<(tiles + 3) / 4, 128, 0, stream>>>(x, W1, bufA, N, FIN, FIN);
  }
  alphas_kernel<<<nodeWaveBlk, 256, 0, stream>>>(bufA, aS1, aD1, as1, ad1, N, H1, D);
  edge_max_kernel<<<edgeBlocks, 256, 0, stream>>>(ei, E, N, as1, ad1, m1, H1);
  edge_expsum_kernel<<<edgeBlocks, 256, 0, stream>>>(ei, E, N, as1, ad1, m1, s1, H1);
  edge_agg_kernel<<<edgeWaveBlk, 256, 0, stream>>>(ei, E, N, bufA, as1, ad1, m1, s1,
                                                   bufB, H1, D);
  finalize_kernel<<<ndBlocks, 256, 0, stream>>>(bufB, b1, bufC, N, H1, D);

  // ================= Layer 2 (heads=1, D=128) =================
  hipMemsetAsync(bufB, 0, (size_t)N * D * 4, stream);    // agg2 (after agg1 consumed)
  {
    const int tiles = ((N + 15) / 16) * (D / 16);
    gemm_f32_wmma<<<(tiles + 3) / 4, 128, 0, stream>>>(bufC, W2, bufA, N, D, D);
  }
  alphas_kernel<<<nodeWaveBlk, 256, 0, stream>>>(bufA, aS2, aD2, as2, ad2, N, 1, D);
  edge_max_kernel<<<edgeBlocks, 256, 0, stream>>>(ei, E, N, as2, ad2, m2, 1);
  edge_expsum_kernel<<<edgeBlocks, 256, 0, stream>>>(ei, E, N, as2, ad2, m2, s2, 1);
  edge_agg_kernel<<<edgeWaveBlk, 256, 0, stream>>>(ei, E, N, bufA, as2, ad2, m2, s2,
                                                   bufB, 1, D);
  finalize_kernel<<<ndBlocks, 256, 0, stream>>>(bufB, b2, bufC, N, 1, D);

  // ================= Pool + MLP head =================
  pool_kernel<<<ndBlocks, 256, 0, stream>>>(bufC, batch, sums, cnt, N, D);
  head_kernel<<<G, 128, 0, stream>>>(sums, cnt, Wc1, bc1, Wc2, bc2, out, D);
}